// Model_21835613732962
// MI455X (gfx1250) — compile-verified
//
#include <hip/hip_runtime.h>
#include <hip/hip_bf16.h>
#include <math.h>

typedef __attribute__((ext_vector_type(2))) float v2f;
typedef __attribute__((ext_vector_type(8))) float v8f;

#define DD 512          // feature dim D
#define KIN 3072        // C*H*W
#define MS 8192         // B*P rows of x_s
#define MT 128          // B rows of x_t
#define NBATCH 128
#define PP 64
#define ASTRIDE 20      // padded LDS row stride (dwords): 16B-aligned rows, conflict-free
#define DISC_BLOCKS 2064   // (8192 + 8192 + 128) rows / 8 waves per block

__device__ __forceinline__ float lsig(float z) {
    // log(sigmoid(z)) = min(z,0) - log1p(exp(-|z|))  (stable)
    return fminf(z, 0.0f) - log1pf(expf(-fabsf(z)));
}

// CDNA5 async global->LDS copy, 16 bytes per lane. Tracked by ASYNCcnt.
// LDS destination = low 32 bits of the generic pointer (DS offset lives in addr[31:0]).
__device__ __forceinline__ void async_copy16(const void* gptr, void* lptr) {
    unsigned loff = (unsigned)(uintptr_t)lptr;
    asm volatile("global_load_async_to_lds_b128 %0, %1, off"
                 :: "v"(loff), "v"(gptr) : "memory");
}
__device__ __forceinline__ void wait_async0() {
    asm volatile("s_wait_asynccnt 0x0" ::: "memory");
}

// -----------------------------------------------------------------------------
// OUT[row, :] = act(Arow @ B[K,N] + bias),  row < M0 -> A0/O0, else A1/O1.
// 64x64 tile per block, 256 threads = 8 wave32, each wave: two 16x16 WMMA accs.
// V_WMMA_F32_16X16X4_F32 (fp32 in / fp32 acc).
// Double-buffered LDS staged via GLOBAL_LOAD_ASYNC_TO_LDS_B128: prefetch of
// chunk kc+16 overlaps the 8 WMMAs of chunk kc; s_wait_asynccnt after compute.
// A frag layout (ISA 7.12.2, 32-bit A 16x4): lane<16: K=k,k+1 ; lane>=16: K=k+2,k+3
// -----------------------------------------------------------------------------
__global__ __launch_bounds__(256) void gemm_relu_wmma(
    const float* __restrict__ A0, unsigned long long M0,
    const float* __restrict__ A1,
    const float* __restrict__ B,
    const float* __restrict__ bias,
    float* __restrict__ O0,
    float* __restrict__ O1,
    int K, int N, int do_relu)
{
    __shared__ float As[2][64 * ASTRIDE];
    __shared__ float Bs[2][16 * 64];

    const int tid  = threadIdx.x;
    const int wave = tid >> 5;
    const int lane = tid & 31;
    const int hl   = lane >> 4;     // half of wave (K-split)
    const int m    = lane & 15;     // M (or N) index within 16
    const int r    = wave & 3;      // M-subtile 0..3
    const int nOff = (wave >> 2) * 32;  // N-subtile base (each wave does nOff, nOff+16)

    const size_t mBase = (size_t)blockIdx.y * 64;
    const int    nBase = blockIdx.x * 64;

    v8f acc0 = {0,0,0,0,0,0,0,0};
    v8f acc1 = {0,0,0,0,0,0,0,0};

    // A-tile stage: 64 rows x 16 k, one async b128 per thread
    const int arow = tid >> 2;
    const int acol = (tid & 3) << 2;
    const size_t grow = mBase + (size_t)arow;
    const float* aptr = (grow < M0) ? (A0 + grow * (size_t)K)
                                    : (A1 + (grow - M0) * (size_t)K);
    // B-tile stage: 16 k x 64 n, one async b128 per thread
    const int brow = tid >> 4;
    const int bcol = (tid & 15) << 2;
    const float* bptr = B + (size_t)brow * N + nBase + bcol;

    // Prologue: stage chunk 0
    async_copy16(aptr + acol,  &As[0][arow * ASTRIDE + acol]);
    async_copy16(bptr,         &Bs[0][brow * 64 + bcol]);
    wait_async0();
    __syncthreads();

    int cur = 0;
    for (int kc = 0; kc < K; kc += 16) {
        const int nxt = cur ^ 1;
        if (kc + 16 < K) {   // prefetch next chunk while computing this one
            async_copy16(aptr + kc + 16 + acol,
                         &As[nxt][arow * ASTRIDE + acol]);
            async_copy16(bptr + (size_t)(kc + 16) * N,
                         &Bs[nxt][brow * 64 + bcol]);
        }
        const float* __restrict__ Ac = As[cur];
        const float* __restrict__ Bc = Bs[cur];
        #pragma unroll
        for (int k0 = 0; k0 < 16; k0 += 4) {
            const int ka = k0 + 2 * hl;
            v2f a, b0, b1;
            a.x  = Ac[(r * 16 + m) * ASTRIDE + ka];      // ds_load_b64 (8B aligned)
            a.y  = Ac[(r * 16 + m) * ASTRIDE + ka + 1];
            b0.x = Bc[ka * 64 + nOff + m];
            b0.y = Bc[(ka + 1) * 64 + nOff + m];
            b1.x = Bc[ka * 64 + nOff + 16 + m];
            b1.y = Bc[(ka + 1) * 64 + nOff + 16 + m];
            acc0 = __builtin_amdgcn_wmma_f32_16x16x4_f32(false, a, false, b0,
                                                         (short)0, acc0, false, false);
            acc1 = __builtin_amdgcn_wmma_f32_16x16x4_f32(false, a, false, b1,
                                                         (short)0, acc1, false, false);
        }
        wait_async0();       // prefetch landed (overlapped with the WMMAs above)
        __syncthreads();
        cur = nxt;
    }

    // Epilogue: C/D layout — VGPR i: lanes0-15 M=i, lanes16-31 M=8+i; N=lane&15
    #pragma unroll
    for (int i = 0; i < 8; ++i) {
        const size_t row = mBase + (size_t)(r * 16 + hl * 8 + i);
        float* out = (row < M0) ? (O0 + row * (size_t)N)
                                : (O1 + (row - M0) * (size_t)N);
        const int c0 = nBase + nOff + m;
        float v0 = acc0[i] + bias[c0];
        float v1 = acc1[i] + bias[c0 + 16];
        if (do_relu) { v0 = fmaxf(v0, 0.0f); v1 = fmaxf(v1, 0.0f); }
        out[c0]      = v0;
        out[c0 + 16] = v1;
    }
}

// -----------------------------------------------------------------------------
// Per-batch verify loss: G = F[64,512] @ F^T ; p_b = sum_i rowsum_i/(G_ii+eps)
// One block per batch. B-fragment is read from the same LDS F tile (B = F^T).
// Same async double-buffered staging as the GEMM.
// -----------------------------------------------------------------------------
__global__ __launch_bounds__(256) void gram_verify_wmma(
    const float* __restrict__ Fin, float* __restrict__ out_p)
{
    __shared__ float Fs[2][64 * ASTRIDE];
    __shared__ float Gs[64 * 65];
    __shared__ float ps[64];

    const int tid  = threadIdx.x;
    const int wave = tid >> 5;
    const int lane = tid & 31;
    const int hl   = lane >> 4;
    const int m    = lane & 15;
    const int r    = wave & 3;
    const int nOff = (wave >> 2) * 32;

    const float* F = Fin + (size_t)blockIdx.x * PP * DD;

    v8f acc0 = {0,0,0,0,0,0,0,0};
    v8f acc1 = {0,0,0,0,0,0,0,0};

    const int arow = tid >> 2;
    const int acol = (tid & 3) << 2;
    const float* fptr = F + (size_t)arow * DD + acol;

    async_copy16(fptr, &Fs[0][arow * ASTRIDE + acol]);
    wait_async0();
    __syncthreads();

    int cur = 0;
    for (int kc = 0; kc < DD; kc += 16) {
        const int nxt = cur ^ 1;
        if (kc + 16 < DD) {
            async_copy16(fptr + kc + 16, &Fs[nxt][arow * ASTRIDE + acol]);
        }
        const float* __restrict__ Fc = Fs[cur];
        #pragma unroll
        for (int k0 = 0; k0 < 16; k0 += 4) {
            const int ka = k0 + 2 * hl;
            v2f a, b0, b1;
            a.x  = Fc[(r * 16 + m) * ASTRIDE + ka];
            a.y  = Fc[(r * 16 + m) * ASTRIDE + ka + 1];
            b0.x = Fc[(nOff + m) * ASTRIDE + ka];        // B[k][n] = F[n][k]
            b0.y = Fc[(nOff + m) * ASTRIDE + ka + 1];
            b1.x = Fc[(nOff + 16 + m) * ASTRIDE + ka];
            b1.y = Fc[(nOff + 16 + m) * ASTRIDE + ka + 1];
            acc0 = __builtin_amdgcn_wmma_f32_16x16x4_f32(false, a, false, b0,
                                                         (short)0, acc0, false, false);
            acc1 = __builtin_amdgcn_wmma_f32_16x16x4_f32(false, a, false, b1,
                                                         (short)0, acc1, false, false);
        }
        wait_async0();
        __syncthreads();
        cur = nxt;
    }

    #pragma unroll
    for (int i = 0; i < 8; ++i) {
        const int row = r * 16 + hl * 8 + i;
        Gs[row * 65 + nOff + m]      = acc0[i];
        Gs[row * 65 + nOff + 16 + m] = acc1[i];
    }
    __syncthreads();

    if (tid < 64) {
        float rowsum = 0.0f;
        #pragma unroll 8
        for (int t = 0; t < 64; ++t) rowsum += Gs[tid * 65 + t];
        ps[tid] = rowsum / (Gs[tid * 65 + tid] + 1e-6f);
    }
    __syncthreads();
    if (tid == 0) {
        float s = 0.0f;
        for (int i = 0; i < 64; ++i) s += ps[i];   // fixed order -> deterministic
        out_p[blockIdx.x] = s;
    }
}

// -----------------------------------------------------------------------------
// Discriminator heads: one wave per row; rows = [f_s:8192 | g_s:8192 | f_t:128]
// f_s rows  -> p1 += ls(z1), p2 += ls(z2), p3 += ls(-z2)
// g_s rows  -> p1 += ls(-z2)
// f_t rows  -> p1 += ls(-z2)
// Category boundaries (8192, 128) are multiples of 8 rows/block.
// -----------------------------------------------------------------------------
__global__ __launch_bounds__(256) void disc_reduce(
    const float* __restrict__ fs, const float* __restrict__ gs,
    const float* __restrict__ ft,
    const float* __restrict__ wd1, const float* __restrict__ bd1,
    const float* __restrict__ wd2, const float* __restrict__ bd2,
    float* __restrict__ p1, float* __restrict__ p2, float* __restrict__ p3)
{
    __shared__ float s1[8], s2[8], s3[8];
    const int tid  = threadIdx.x;
    const int wave = tid >> 5;
    const int lane = tid & 31;
    const size_t row = (size_t)blockIdx.x * 8 + wave;

    const float* F;
    int cat;
    if (row < MS)            { F = fs + row * DD;               cat = 0; }
    else if (row < 2 * MS)   { F = gs + (row - MS) * DD;        cat = 1; }
    else                     { F = ft + (row - 2 * MS) * DD;    cat = 2; }

    float d1 = 0.0f, d2 = 0.0f;
    for (int e = lane; e < DD; e += 32) {
        const float fv = F[e];
        d1 += fv * wd1[e];
        d2 += fv * wd2[e];
    }
    #pragma unroll
    for (int off = 16; off > 0; off >>= 1) {
        d1 += __shfl_xor(d1, off, 32);
        d2 += __shfl_xor(d2, off, 32);
    }
    if (lane == 0) {
        if (cat == 0) {
            const float z1 = d1 + bd1[0];
            const float z2 = d2 + bd2[0];
            s1[wave] = lsig(z1);
            s2[wave] = lsig(z2);
            s3[wave] = lsig(-z2);
        } else {
            const float z2 = d2 + bd2[0];
            s1[wave] = lsig(-z2);
            s2[wave] = 0.0f;
            s3[wave] = 0.0f;
        }
    }
    __syncthreads();
    if (tid == 0) {
        float a = 0.0f, b = 0.0f, c = 0.0f;
        for (int i = 0; i < 8; ++i) { a += s1[i]; b += s2[i]; c += s3[i]; }
        p1[blockIdx.x] = a;
        p2[blockIdx.x] = b;
        p3[blockIdx.x] = c;
    }
}

// -----------------------------------------------------------------------------
// Final fixed-order fold into the scalar loss.
// loss = 0.5*vrf_f + 1.5*vrf_g + S1 + 0.5*S2 + 0.25*S3 + 0.25*S4 + 0.5*S5
// -----------------------------------------------------------------------------
__global__ void finalize_loss(
    const float* __restrict__ pf, const float* __restrict__ pg,
    const float* __restrict__ p1, const float* __restrict__ p2,
    const float* __restrict__ p3, float* __restrict__ out)
{
    if (threadIdx.x != 0 || blockIdx.x != 0) return;
    float S1 = 0.0f, S2 = 0.0f, S5 = 0.0f, S3 = 0.0f, S4 = 0.0f;
    for (int i = 0; i < 1024; ++i) { S1 += p1[i]; S2 += p2[i]; S5 += p3[i]; }
    for (int i = 1024; i < 2048; ++i) S3 += p1[i];
    for (int i = 2048; i < DISC_BLOCKS; ++i) S4 += p1[i];
    float vf = 0.0f, vg = 0.0f;
    for (int b = 0; b < NBATCH; ++b) { vf += pf[b]; vg += pg[b]; }
    vf /= (float)NBATCH;  vg /= (float)NBATCH;
    S1 /= (float)MS; S2 /= (float)MS; S5 /= (float)MS; S3 /= (float)MS;
    S4 /= (float)MT;
    out[0] = 0.5f * vf + 1.5f * vg + S1 + 0.5f * S2 + 0.25f * S3 + 0.25f * S4 + 0.5f * S5;
}

extern "C" void kernel_launch(void* const* d_in, const int* in_sizes, int n_in,
                              void* d_out, int out_size, void* d_ws, size_t ws_size,
                              hipStream_t stream) {
    (void)in_sizes; (void)n_in; (void)out_size; (void)ws_size;

    const float* x_s = (const float*)d_in[0];   // [128,64,3,32,32] -> [8192,3072]
    const float* x_t = (const float*)d_in[1];   // [128,3,32,32]    -> [128,3072]
    /* d_in[2] = y_s (int) — unused by the loss */
    const float* Wg  = (const float*)d_in[3];   // [3072,512]
    const float* bg  = (const float*)d_in[4];   // [512]
    const float* Wt  = (const float*)d_in[5];   // [512,512]
    const float* bt  = (const float*)d_in[6];   // [512]
    const float* wd1 = (const float*)d_in[7];   // [512]
    const float* bd1 = (const float*)d_in[8];   // scalar
    const float* wd2 = (const float*)d_in[9];   // [512]
    const float* bd2 = (const float*)d_in[10];  // scalar

    // Workspace layout (floats): fully overwritten every call -> deterministic.
    float* f_s   = (float*)d_ws;                    // 8192*512
    float* g_s   = f_s + (size_t)MS * DD;           // 8192*512
    float* f_t   = g_s + (size_t)MS * DD;           // 128*512
    float* pf    = f_t + (size_t)MT * DD;           // 128
    float* pg    = pf + NBATCH;                     // 128
    float* part1 = pg + NBATCH;                     // 2064
    float* part2 = part1 + DISC_BLOCKS;             // 2064
    float* part3 = part2 + DISC_BLOCKS;             // 2064

    // 1) f = relu([x_s;x_t] @ Wg + bg)   (M = 8320 = 130 tiles of 64)
    gemm_relu_wmma<<<dim3(DD / 64, (MS + MT) / 64), 256, 0, stream>>>(
        x_s, (unsigned long long)MS, x_t, Wg, bg, f_s, f_t, KIN, DD, 1);

    // 2) g_s = f_s @ Wt + bt
    gemm_relu_wmma<<<dim3(DD / 64, MS / 64), 256, 0, stream>>>(
        f_s, (unsigned long long)MS, f_s, Wt, bt, g_s, g_s, DD, DD, 0);

    // 3) verify terms (Gram matrices) for f_s and g_s
    gram_verify_wmma<<<NBATCH, 256, 0, stream>>>(f_s, pf);
    gram_verify_wmma<<<NBATCH, 256, 0, stream>>>(g_s, pg);

    // 4) discriminator heads + log-sigmoid partials
    disc_reduce<<<DISC_BLOCKS, 256, 0, stream>>>(
        f_s, g_s, f_t, wd1, bd1, wd2, bd2, part1, part2, part3);

    // 5) fold into scalar loss
    finalize_loss<<<1, 64, 0, stream>>>(pf, pg, part1, part2, part3, (float*)d_out);
}